// SupplyGraphModel_41549513621817
// MI455X (gfx1250) — compile-verified
//
#include <hip/hip_runtime.h>

typedef float v2f __attribute__((ext_vector_type(2)));
typedef float v8f __attribute__((ext_vector_type(8)));

#define NN    50000
#define NRELS 3
#define NE    640000
#define D     128
#define DOUT  32

// ---------------------------------------------------------------- zero
__global__ void zero_kernel(float* __restrict__ p, int n) {
  int i = blockIdx.x * blockDim.x + threadIdx.x;
  int stride = gridDim.x * blockDim.x;
  for (; i < n; i += stride) p[i] = 0.0f;
}

// ------------------------------------------------- fold root weights/bias
// WrSum = Wr[l,0]+Wr[l,1]+Wr[l,2]  (128x128), bSum = b[l,0]+b[l,1]+b[l,2]
__global__ void wsum_kernel(const float* __restrict__ Wr, const float* __restrict__ b,
                            float* __restrict__ WrSum, float* __restrict__ bSum) {
  int i = blockIdx.x * blockDim.x + threadIdx.x;
  if (i < D * D) WrSum[i] = Wr[i] + Wr[D * D + i] + Wr[2 * D * D + i];
  if (i < D)     bSum[i]  = b[i] + b[D + i] + b[2 * D + i];
}

// ------------------------------------------------- edge scatter (segment sum)
// one wave per edge; each lane gathers float4 of h[src] and atomically adds
// into summed[dst]; lane 0 bumps the degree counter. All hot data (~51 MB)
// is L2-resident (192 MB L2) so this is L2 atomic-throughput bound.
__global__ void scatter_kernel(const float* __restrict__ h, const int* __restrict__ src,
                               const int* __restrict__ dst, float* __restrict__ summed,
                               float* __restrict__ deg, int E) {
  int t = blockIdx.x * blockDim.x + threadIdx.x;
  int edge = t >> 5;
  int lane = t & 31;
  if (edge >= E) return;
  int s = src[edge];
  int d = dst[edge];
  float4 v = ((const float4*)(h + (size_t)s * D))[lane];
  float* o = summed + (size_t)d * D + lane * 4;
  atomicAdd(o + 0, v.x);
  atomicAdd(o + 1, v.y);
  atomicAdd(o + 2, v.z);
  atomicAdd(o + 3, v.w);
  if (lane == 0) atomicAdd(deg + d, 1.0f);
}

// ------------------------------------------------- WMMA fp32 GEMM
// C[M,No] = (beta ? C : bias) + scale(A)[M,128] @ B[128,No], optional ReLU.
// scale(A): if degp != nullptr, row m of A is multiplied by 1/max(deg[m],1)
// (SAGE mean aggregation) while resident in LDS -- avoids a 51 MB global pass.
// A tile (8 KB, contiguous) is staged via CDNA5 GLOBAL_LOAD_ASYNC_TO_LDS_B128
// (ASYNCcnt path); B is staged transposed through VGPRs.
// grid.x = M/16 node tiles; (No/16) waves of 32 per block; inner loop is
// 32x v_wmma_f32_16x16x4_f32 fed by ds_load_b64.
__global__ void gemm_wmma_kernel(const float* __restrict__ A, const float* __restrict__ B,
                                 float* __restrict__ C, const float* __restrict__ bias,
                                 const float* __restrict__ degp,
                                 int No, int beta, int relu) {
  extern __shared__ float lds[];
  float* bT = lds;             // [No][128]  B transposed (K pairs contiguous)
  float* aS = lds + No * D;    // [16][128]
  const int tid = threadIdx.x;
  const int nt  = blockDim.x;
  const int m0  = blockIdx.x * 16;

  // A tile: contiguous 16x128 floats -> async global->LDS, 16B per issue.
  // Low 32 bits of the generic shared-mem pointer == LDS allocation offset.
  for (int j = tid * 4; j < 16 * D; j += nt * 4) {
    unsigned lofs  = (unsigned)(unsigned long long)(aS + j);
    unsigned long long gaddr = (unsigned long long)(A + (size_t)m0 * D + j);
    asm volatile("global_load_async_to_lds_b128 %0, %1, off"
                 :: "v"(lofs), "v"(gaddr) : "memory");
  }

  for (int i = tid; i < No * D; i += nt) {  // coalesced row-major read of B
    int k = i / No, n = i % No;
    bT[n * D + k] = B[i];
  }

  asm volatile("s_wait_asynccnt 0x0" ::: "memory");
  __syncthreads();

  if (degp) {  // fused mean: scale staged rows by reciprocal degree
    for (int i = tid; i < 16 * D; i += nt) {
      float dg = degp[m0 + (i >> 7)];
      aS[i] *= 1.0f / fmaxf(dg, 1.0f);
    }
    __syncthreads();
  }

  const int wave = tid >> 5;
  const int lane = tid & 31;
  const int lm   = lane & 15;   // M (for A) / N (for B,C) within tile
  const int hi   = lane >> 4;   // selects K pair (A/B) and M half (C/D)
  const int nb   = wave * 16;

  v8f c;
  if (beta) {
    #pragma unroll
    for (int j = 0; j < 8; ++j)
      c[j] = C[(size_t)(m0 + j + 8 * hi) * No + nb + lm];
  } else {
    float bv = bias ? bias[nb + lm] : 0.0f;
    #pragma unroll
    for (int j = 0; j < 8; ++j) c[j] = bv;
  }

  #pragma unroll
  for (int k = 0; k < D; k += 4) {
    int ka = k + hi * 2;
    v2f a = *(const v2f*)(aS + lm * D + ka);         // A[m][ka], A[m][ka+1]
    v2f b = *(const v2f*)(bT + (nb + lm) * D + ka);  // B[ka][n], B[ka+1][n]
    c = __builtin_amdgcn_wmma_f32_16x16x4_f32(false, a, false, b,
                                              (short)0, c, false, false);
  }

  #pragma unroll
  for (int j = 0; j < 8; ++j) {
    float v = c[j];
    if (relu) v = fmaxf(v, 0.0f);
    C[(size_t)(m0 + j + 8 * hi) * No + nb + lm] = v;
  }
}

// ----------------------------------------------------------------
extern "C" void kernel_launch(void* const* d_in, const int* in_sizes, int n_in,
                              void* d_out, int out_size, void* d_ws, size_t ws_size,
                              hipStream_t stream) {
  const float* x     = (const float*)d_in[0];
  const int*   edges = (const int*)d_in[1];   // [3][2][E]
  const float* Wn    = (const float*)d_in[2]; // [2][3][128][128]
  const float* Wr    = (const float*)d_in[3]; // [2][3][128][128]
  const float* bc    = (const float*)d_in[4]; // [2][3][128]
  const float* linW  = (const float*)d_in[5]; // [128][32]
  const float* linb  = (const float*)d_in[6]; // [32]
  float* out = (float*)d_out;                 // [50000][32]

  const size_t HD = (size_t)NN * D;           // 6.4M floats
  float* ws    = (float*)d_ws;
  float* hbuf0 = ws;
  float* hbuf1 = hbuf0 + HD;
  float* sumb  = hbuf1 + HD;                  // raw segment sums
  float* deg   = sumb + HD;                   // contiguous: zero both in one pass
  float* WrSum = deg + NN;
  float* bSum  = WrSum + D * D;

  const int    mt      = NN / 16;                                   // 3125
  const size_t ldsFull = (size_t)(D * D + 16 * D) * sizeof(float);  // 72 KB
  const size_t ldsOut  = (size_t)(DOUT * D + 16 * D) * sizeof(float);

  const float* hin  = x;
  float*       hout = hbuf1;
  for (int l = 0; l < 2; ++l) {
    // acc = h @ sum_r Wr[l,r] + sum_r b[l,r]
    wsum_kernel<<<(D * D + 255) / 256, 256, 0, stream>>>(
        Wr + (size_t)l * NRELS * D * D, bc + (size_t)l * NRELS * D, WrSum, bSum);
    gemm_wmma_kernel<<<mt, 256, ldsFull, stream>>>(hin, WrSum, hout, bSum,
                                                   nullptr, D, 0, 0);

    for (int r = 0; r < NRELS; ++r) {
      zero_kernel<<<2048, 256, 0, stream>>>(sumb, (int)(HD + NN));
      scatter_kernel<<<(NE * 32) / 256, 256, 0, stream>>>(
          hin, edges + ((size_t)r * 2 + 0) * NE, edges + ((size_t)r * 2 + 1) * NE,
          sumb, deg, NE);
      // acc += (sum/deg) @ Wn[l,r]; mean fused into A staging; ReLU on last rel
      gemm_wmma_kernel<<<mt, 256, ldsFull, stream>>>(
          sumb, Wn + (size_t)(l * NRELS + r) * D * D, hout, nullptr, deg, D, 1,
          (r == NRELS - 1) ? 1 : 0);
    }
    hin  = hout;
    hout = (l == 0) ? hbuf0 : hbuf1;
  }

  // out = h @ lin_W + lin_b   (No = 32 -> 2 waves per block)
  gemm_wmma_kernel<<<mt, 64, ldsOut, stream>>>(hin, linW, out, linb,
                                               nullptr, DOUT, 0, 0);
}